// LSTMTrajectoryEncoder_71459665871186
// MI455X (gfx1250) — compile-verified
//
#include <hip/hip_runtime.h>
#include <math.h>

// ---------------------------------------------------------------------------
// GRU trajectory encoder for MI455X (gfx1250), fp32 via V_WMMA_F32_16X16X4_F32
// + async global->LDS double-buffered gate streaming (ASYNCcnt)
// T=1000, B=256, STATE=64, ACTION=8, REWARD=1, EMBED=64, HIDDEN=256
// ---------------------------------------------------------------------------

#define TT 1000
#define BB 256
#define HH 256     // hidden
#define GD 768     // 3*HIDDEN gate width
#define XD 192     // 3*EMBED input width

typedef float v2f __attribute__((ext_vector_type(2)));
typedef float v8f __attribute__((ext_vector_type(8)));
typedef int   v4i __attribute__((vector_size(16)));   // matches builtin proto

#if defined(__has_builtin)
#if __has_builtin(__builtin_amdgcn_global_load_async_to_lds_b128)
#define HAVE_ASYNC_LDS 1
#endif
#endif

// D = A(16x4 f32) * B(4x16 f32) + C(16x16 f32)
__device__ __forceinline__ v8f wmma_f32_16x16x4(v2f a, v2f b, v8f c) {
  return __builtin_amdgcn_wmma_f32_16x16x4_f32(
      /*neg_a=*/false, a, /*neg_b=*/false, b,
      /*c_mod=*/(short)0, c, /*reuse_a=*/false, /*reuse_b=*/false);
}

#ifdef HAVE_ASYNC_LDS
__device__ __forceinline__ void async_copy16(const float* g, float* l) {
  __builtin_amdgcn_global_load_async_to_lds_b128(
      (__attribute__((address_space(1))) v4i*)(const_cast<float*>(g)),
      (__attribute__((address_space(3))) v4i*)l, /*offset=*/0, /*cpol=*/0);
}
__device__ __forceinline__ void wait_async0() {
#if __has_builtin(__builtin_amdgcn_s_wait_asynccnt)
  __builtin_amdgcn_s_wait_asynccnt(0);
#else
  asm volatile("s_wait_asynccnt 0" ::: "memory");
#endif
}
#else
__device__ __forceinline__ void wait_async0() {}
#endif

// copy one 16x768 gate tile (12288 floats, contiguous) global -> LDS
__device__ __forceinline__ void issue_gtile(const float* __restrict__ g,
                                            float* __restrict__ dst, int tid) {
  for (int i = tid * 4; i < 16 * GD; i += 512 * 4) {
#ifdef HAVE_ASYNC_LDS
    async_copy16(g + i, dst + i);
#else
    *(float4*)&dst[i] = *(const float4*)&g[i];
#endif
  }
}

__device__ __forceinline__ float gelu_tanh(float x) {
  // jax.nn.gelu default (approximate=True)
  const float c0 = 0.7978845608028654f;  // sqrt(2/pi)
  const float c1 = 0.044715f;
  float x3 = x * x * x;
  return 0.5f * x * (1.0f + tanhf(c0 * (x + c1 * x3)));
}

__device__ __forceinline__ float sigmoidf_fast(float x) {
  return 1.0f / (1.0f + __expf(-x));
}

// ---------------------------------------------------------------------------
// Kernel 0 (workspace path): interleave K-pairs of wi/wh so each lane's
// B-fragment (rows k,k+1 at column n) is one contiguous float2 -> 1 b64 load.
// packed[(k/2)*2*N + n*2 + (k&1)] = w[k*N + n]
// ---------------------------------------------------------------------------
__global__ __launch_bounds__(256)
void pack_weights_kernel(const float* __restrict__ wi, const float* __restrict__ wh,
                         float* __restrict__ wip, float* __restrict__ whp)
{
  const int stride = gridDim.x * blockDim.x;
  for (int i = blockIdx.x * blockDim.x + threadIdx.x; i < XD * GD; i += stride) {
    const int k = i / GD, n = i % GD;
    wip[(size_t)(k >> 1) * (2 * GD) + n * 2 + (k & 1)] = wi[i];
  }
  for (int i = blockIdx.x * blockDim.x + threadIdx.x; i < HH * GD; i += stride) {
    const int k = i / GD, n = i % GD;
    whp[(size_t)(k >> 1) * (2 * GD) + n * 2 + (k & 1)] = wh[i];
  }
}

// ---------------------------------------------------------------------------
// Kernel 1 (workspace path): fused embeddings + gx = gelu(embed) @ wi + bg
// Grid: (T*B)/64 blocks of 256 threads (8 waves). Uses packed wi.
// ---------------------------------------------------------------------------
__global__ __launch_bounds__(256)
void embed_gx_kernel(const float* __restrict__ states, const float* __restrict__ actions,
                     const float* __restrict__ rewards,
                     const float* __restrict__ Ws, const float* __restrict__ bs,
                     const float* __restrict__ Wa, const float* __restrict__ ba,
                     const float* __restrict__ Wr, const float* __restrict__ br,
                     const float* __restrict__ wip, const float* __restrict__ bg,
                     float* __restrict__ gx)
{
  __shared__ __align__(16) float sx[64 * 196];  // 64 rows x 192 (stride 196: 4-bank skew)

  const int tid   = threadIdx.x;
  const int wave  = tid >> 5;
  const int lane  = tid & 31;
  const int lrow  = lane & 15;
  const int khalf = lane >> 4;            // 0: K pair {0,1}, 1: K pair {2,3}
  const size_t row0 = (size_t)blockIdx.x * 64;   // flat (t*B + b) row base

  // ---- embeddings: x = [gelu(s@Ws+bs) | gelu(a@Wa+ba) | gelu(r@Wr+br)] ----
  for (int i = tid; i < 64 * 64; i += 256) {
    const int m = i >> 6, c = i & 63;
    const float* srow = states + (row0 + m) * 64;
    float a0 = bs[c];
    #pragma unroll 8
    for (int k = 0; k < 64; ++k) a0 += srow[k] * Ws[k * 64 + c];
    sx[m * 196 + c] = gelu_tanh(a0);

    const float* arow = actions + (row0 + m) * 8;
    float a1 = ba[c];
    #pragma unroll
    for (int k = 0; k < 8; ++k) a1 += arow[k] * Wa[k * 64 + c];
    sx[m * 196 + 64 + c] = gelu_tanh(a1);

    float a2 = br[c] + rewards[row0 + m] * Wr[c];
    sx[m * 196 + 128 + c] = gelu_tanh(a2);
  }
  __syncthreads();

  // ---- gx tile = x @ wi + bg via WMMA: 4 M-subtiles x 48 N-tiles over 8 waves
  for (int tileidx = wave; tileidx < 4 * 48; tileidx += 8) {
    const int mt = tileidx / 48;
    const int nt = tileidx % 48;
    const int n0 = nt * 16;
    const int mrow = mt * 16 + lrow;
    v8f acc = {};
    const float* Wb = wip + (size_t)khalf * (2 * GD) + (size_t)(n0 + lrow) * 2;
    #pragma unroll 4
    for (int k0 = 0; k0 < XD; k0 += 4) {
      v2f av = *(const v2f*)&sx[mrow * 196 + k0 + 2 * khalf];
      v2f bv = *(const v2f*)(Wb + (size_t)k0 * GD);
      acc = wmma_f32_16x16x4(av, bv, acc);
    }
    const float bgv = bg[n0 + lrow];
    float* g = gx + (row0 + (size_t)mt * 16) * GD + n0 + lrow;
    #pragma unroll
    for (int i = 0; i < 8; ++i)
      g[(size_t)(i + 8 * khalf) * GD] = acc[i] + bgv;
  }
}

// ---------------------------------------------------------------------------
// Kernel 2: persistent GRU scan. 16 blocks x 512 threads (16 waves).
// Block bt owns batch rows [16*bt, 16*bt+16); h tile lives in LDS for all T.
//   zr = sigmoid(G[:, :512] + h @ wh[:, :512])
//   a  = tanh  (G[:, 512:] + (r*h) @ wh[:, 512:])
//   h  = (1-z)*h + z*a
// USE_GX=true : G async-double-buffered from precomputed gx; packed wh.
// USE_GX=false: G computed in-loop (embed + x@wi via WMMA), zero workspace.
// ---------------------------------------------------------------------------
template <bool USE_GX>
__global__ __launch_bounds__(512)
void gru_scan_kernel(const float* __restrict__ gx, const float* __restrict__ whp,
                     const float* __restrict__ states, const float* __restrict__ actions,
                     const float* __restrict__ rewards,
                     const float* __restrict__ Ws, const float* __restrict__ bs,
                     const float* __restrict__ Wa, const float* __restrict__ ba,
                     const float* __restrict__ Wr, const float* __restrict__ br,
                     const float* __restrict__ wi, const float* __restrict__ wh,
                     const float* __restrict__ bg,
                     float* __restrict__ out)
{
  __shared__ __align__(16) float sh_h [16 * 260];    // h tile, stride 260 (bank skew)
  __shared__ __align__(16) float sh_rh[16 * 260];    // r*h tile
  __shared__ __align__(16) float sh_G [2 * 16 * GD]; // double-buffered gate tiles
  __shared__ __align__(16) float sh_x [16 * 196];    // x tile (fused path only)

  const int tid   = threadIdx.x;
  const int wave  = tid >> 5;
  const int lane  = tid & 31;
  const int lrow  = lane & 15;
  const int khalf = lane >> 4;
  const int r0    = blockIdx.x * 16;   // batch row base

  for (int i = tid; i < 16 * 260; i += 512) sh_h[i] = 0.0f;  // h0 = 0
  __syncthreads();

  if (USE_GX) {  // prologue: start streaming gate tile for t=0
    issue_gtile(gx + ((size_t)r0) * GD, sh_G, tid);
  }

  for (int t = 0; t < TT; ++t) {
    const size_t rowbase = (size_t)t * BB + r0;
    float* Gc = USE_GX ? (sh_G + (size_t)(t & 1) * (16 * GD)) : sh_G;

    // ================= make gate pre-activations G available =================
    if (USE_GX) {
      wait_async0();       // my async loads for buf[t&1] have landed
      __syncthreads();     // everyone's loads landed; buf[t&1^1] free to refill
      if (t + 1 < TT)
        issue_gtile(gx + (rowbase + BB) * GD, sh_G + (size_t)((t + 1) & 1) * (16 * GD), tid);
    } else {
      // embeddings for this 16-row tile
      for (int i = tid; i < 16 * 64; i += 512) {
        const int m = i >> 6, c = i & 63;
        const float* srow = states + (rowbase + m) * 64;
        float a0 = bs[c];
        #pragma unroll 8
        for (int k = 0; k < 64; ++k) a0 += srow[k] * Ws[k * 64 + c];
        sh_x[m * 196 + c] = gelu_tanh(a0);

        const float* arow = actions + (rowbase + m) * 8;
        float a1 = ba[c];
        #pragma unroll
        for (int k = 0; k < 8; ++k) a1 += arow[k] * Wa[k * 64 + c];
        sh_x[m * 196 + 64 + c] = gelu_tanh(a1);

        float a2 = br[c] + rewards[rowbase + m] * Wr[c];
        sh_x[m * 196 + 128 + c] = gelu_tanh(a2);
      }
      __syncthreads();
      // G = x @ wi + bg : 48 N-tiles over 16 waves (3 each)
      for (int nt = wave; nt < 48; nt += 16) {
        const int n0 = nt * 16;
        v8f acc = {};
        const float* Wb = wi + (size_t)(2 * khalf) * GD + n0 + lrow;
        #pragma unroll 4
        for (int k0 = 0; k0 < XD; k0 += 4) {
          v2f av = *(const v2f*)&sh_x[lrow * 196 + k0 + 2 * khalf];
          v2f bv;
          bv.x = Wb[(size_t)k0 * GD];
          bv.y = Wb[(size_t)k0 * GD + GD];
          acc = wmma_f32_16x16x4(av, bv, acc);
        }
        const float bgv = bg[n0 + lrow];
        #pragma unroll
        for (int i = 0; i < 8; ++i)
          Gc[(i + 8 * khalf) * GD + n0 + lrow] = acc[i] + bgv;
      }
      __syncthreads();
    }

    // ================= z,r = sigmoid(G_zr + h @ wh_zr) =================
    // N=512 -> 32 tiles over 16 waves (2 each), K=256
    for (int nt = wave; nt < 32; nt += 16) {
      const int n0 = nt * 16;
      v8f acc = {};
      const float* WbP = USE_GX ? (whp + (size_t)khalf * (2 * GD) + (size_t)(n0 + lrow) * 2)
                                : nullptr;
      const float* WbU = wh + (size_t)(2 * khalf) * GD + n0 + lrow;
      #pragma unroll 4
      for (int k0 = 0; k0 < HH; k0 += 4) {
        v2f av = *(const v2f*)&sh_h[lrow * 260 + k0 + 2 * khalf];
        v2f bv;
        if (USE_GX) {
          bv = *(const v2f*)(WbP + (size_t)k0 * GD);
        } else {
          bv.x = WbU[(size_t)k0 * GD];
          bv.y = WbU[(size_t)k0 * GD + GD];
        }
        acc = wmma_f32_16x16x4(av, bv, acc);
      }
      const int col = n0 + lrow;
      #pragma unroll
      for (int i = 0; i < 8; ++i) {
        const int m = i + 8 * khalf;
        Gc[m * GD + col] = sigmoidf_fast(acc[i] + Gc[m * GD + col]);
      }
    }
    __syncthreads();

    // ================= rh = r * h =================
    for (int i = tid; i < 16 * HH; i += 512) {
      const int m = i >> 8, j = i & 255;
      sh_rh[m * 260 + j] = Gc[m * GD + 256 + j] * sh_h[m * 260 + j];
    }
    __syncthreads();

    // ========== a = tanh(G_a + rh @ wh_a); h = (1-z)*h + z*a ==========
    {
      const int n0 = wave * 16;   // 16 tiles, one per wave
      v8f acc = {};
      const float* WbP = USE_GX ? (whp + (size_t)khalf * (2 * GD) + (size_t)(512 + n0 + lrow) * 2)
                                : nullptr;
      const float* WbU = wh + (size_t)(2 * khalf) * GD + 512 + n0 + lrow;
      #pragma unroll 4
      for (int k0 = 0; k0 < HH; k0 += 4) {
        v2f av = *(const v2f*)&sh_rh[lrow * 260 + k0 + 2 * khalf];
        v2f bv;
        if (USE_GX) {
          bv = *(const v2f*)(WbP + (size_t)k0 * GD);
        } else {
          bv.x = WbU[(size_t)k0 * GD];
          bv.y = WbU[(size_t)k0 * GD + GD];
        }
        acc = wmma_f32_16x16x4(av, bv, acc);
      }
      const int col = n0 + lrow;
      float* orow = out + rowbase * HH;
      #pragma unroll
      for (int i = 0; i < 8; ++i) {
        const int m = i + 8 * khalf;
        const float a    = tanhf(acc[i] + Gc[m * GD + 512 + col]);
        const float z    = Gc[m * GD + col];
        const float hold = sh_h[m * 260 + col];
        const float hn   = (1.0f - z) * hold + z * a;
        sh_h[m * 260 + col] = hn;
        orow[(size_t)m * HH + col] = hn;
      }
    }
    __syncthreads();
  }
}

// ---------------------------------------------------------------------------
extern "C" void kernel_launch(void* const* d_in, const int* in_sizes, int n_in,
                              void* d_out, int out_size, void* d_ws, size_t ws_size,
                              hipStream_t stream) {
  (void)in_sizes; (void)n_in; (void)out_size;
  const float* states  = (const float*)d_in[0];
  const float* actions = (const float*)d_in[1];
  const float* rewards = (const float*)d_in[2];
  const float* Ws = (const float*)d_in[3];
  const float* bs = (const float*)d_in[4];
  const float* Wa = (const float*)d_in[5];
  const float* ba = (const float*)d_in[6];
  const float* Wr = (const float*)d_in[7];
  const float* br = (const float*)d_in[8];
  const float* wi = (const float*)d_in[9];
  const float* wh = (const float*)d_in[10];
  const float* bg = (const float*)d_in[11];
  float* out = (float*)d_out;

  const size_t gx_elems  = (size_t)TT * BB * GD;          // 196,608,000 floats
  const size_t wip_elems = (size_t)XD * GD;               // 147,456
  const size_t whp_elems = (size_t)HH * GD;               // 196,608
  const size_t need = (gx_elems + wip_elems + whp_elems) * sizeof(float);  // ~787.8 MB

  if (ws_size >= need) {
    // Path A: pack weights, full-GPU WMMA GEMM for gx, async-streamed scan.
    float* gx  = (float*)d_ws;
    float* wip = gx + gx_elems;
    float* whp = wip + wip_elems;
    pack_weights_kernel<<<576, 256, 0, stream>>>(wi, wh, wip, whp);
    embed_gx_kernel<<<(TT * BB) / 64, 256, 0, stream>>>(
        states, actions, rewards, Ws, bs, Wa, ba, Wr, br, wip, bg, gx);
    gru_scan_kernel<true><<<BB / 16, 512, 0, stream>>>(
        gx, whp, states, actions, rewards, Ws, bs, Wa, ba, Wr, br, wi, wh, bg, out);
  } else {
    // Path B: zero-workspace fused scan (embed + x@wi computed per step).
    gru_scan_kernel<false><<<BB / 16, 512, 0, stream>>>(
        nullptr, nullptr, states, actions, rewards, Ws, bs, Wa, ba, Wr, br, wi, wh, bg, out);
  }
}